// return_pix_86406152061376
// MI455X (gfx1250) — compile-verified
//
#include <hip/hip_runtime.h>

#define Bdim 4
#define Cdim 5
#define Hdim 64
#define Wdim 64
#define HW   (Hdim * Wdim)   // 4096
#define Npts 4096
#define PTILES (HW / 16)     // 256

typedef __attribute__((ext_vector_type(2))) float v2f;
typedef __attribute__((ext_vector_type(8))) float v8f;

// ---------------- K (p-major, padded 8) / V (c-major) maps ----------------
// Kf layout: [b][p][8]  (only slots 0..4 written/read)
// Vf layout: [b][c][p]
__global__ __launch_bounds__(256) void qkv_map_kernel(
    const float* __restrict__ x,
    const float* __restrict__ kw, const float* __restrict__ kb,
    const float* __restrict__ vw, const float* __restrict__ vb,
    float* __restrict__ Kf, float* __restrict__ Vf)
{
    int gid = blockIdx.x * blockDim.x + threadIdx.x;
    if (gid >= Bdim * HW) return;
    int b = gid >> 12;          // / 4096
    int p = gid & (HW - 1);
    float xv[Cdim];
#pragma unroll
    for (int c = 0; c < Cdim; ++c) xv[c] = x[(b * Cdim + c) * HW + p];
#pragma unroll
    for (int o = 0; o < Cdim; ++o) {
        float ka = kb[o], va = vb[o];
#pragma unroll
        for (int c = 0; c < Cdim; ++c) {
            ka = fmaf(kw[o * Cdim + c], xv[c], ka);
            va = fmaf(vw[o * Cdim + c], xv[c], va);
        }
        Kf[((size_t)(b * HW + p)) * 8 + o] = ka;   // p-major
        Vf[(b * Cdim + o) * HW + p]        = va;   // c-major
    }
}

// ---------------- gathered queries: q_n = qw * x[b,:,h,w] + qb ----------------
__global__ __launch_bounds__(256) void q_gather_kernel(
    const float* __restrict__ x,
    const float* __restrict__ qw, const float* __restrict__ qb,
    const int* __restrict__ idx_b, const int* __restrict__ idx_h,
    const int* __restrict__ idx_w, float* __restrict__ Qg)
{
    int n = blockIdx.x * blockDim.x + threadIdx.x;
    if (n >= Npts) return;
    int b   = idx_b[n];
    int pix = idx_h[n] * Wdim + idx_w[n];
    float xv[Cdim];
#pragma unroll
    for (int c = 0; c < Cdim; ++c) xv[c] = x[(b * Cdim + c) * HW + pix];
#pragma unroll
    for (int o = 0; o < Cdim; ++o) {
        float a = qb[o];
#pragma unroll
        for (int c = 0; c < Cdim; ++c) a = fmaf(qw[o * Cdim + c], xv[c], a);
        Qg[n * 8 + o] = a;
    }
    Qg[n * 8 + 5] = 0.0f; Qg[n * 8 + 6] = 0.0f; Qg[n * 8 + 7] = 0.0f;
}

// ---------------- bucket n's by batch ----------------
__global__ void bucket_init_kernel(int* cnts)
{
    if (threadIdx.x < Bdim) cnts[threadIdx.x] = 0;
}

__global__ __launch_bounds__(256) void bucket_kernel(
    const int* __restrict__ idx_b, int* __restrict__ cnts, int* __restrict__ lists)
{
    int n = blockIdx.x * blockDim.x + threadIdx.x;
    if (n >= Npts) return;
    int b   = idx_b[n];
    int pos = atomicAdd(&cnts[b], 1);
    lists[b * Npts + pos] = n;
}

// ---------------- attention: one wave handles 16 queries of one batch ----------------
__global__ __launch_bounds__(32) void attn_kernel(
    const float* __restrict__ Kf, const float* __restrict__ Vf,
    const float* __restrict__ Qg,
    const int* __restrict__ lists, const int* __restrict__ cnts,
    const int* __restrict__ idx_h, const int* __restrict__ idx_w,
    const float* __restrict__ x, const float* __restrict__ gamma,
    float* __restrict__ y)
{
    __shared__ __align__(16) float Etile[16 * 16];  // [p_local][m]

    const int b     = blockIdx.x;
    const int tt    = blockIdx.y;
    const int count = cnts[b];
    if (tt * 16 >= count) return;

    const int lane = threadIdx.x;
    const int half = lane >> 4;     // 0/1
    const int l16  = lane & 15;

    // A-matrix row this lane feeds: m = l16 (clamped-duplicate padding; rows independent)
    int slotA = tt * 16 + l16;
    if (slotA > count - 1) slotA = count - 1;
    const int nA = lists[b * Npts + slotA];

    // A-layout (16x4 f32): v0 holds K=2*half, v1 holds K=2*half+1
    const float2 qp = *(const float2*)(Qg + nA * 8 + 2 * half);   // unconditional b64
    const float  q4 = Qg[nA * 8 + 4];
    v2f aq0, aq1;
    aq0.x = qp.x;  aq0.y = qp.y;
    aq1.x = (half == 0) ? q4 : 0.0f;   // value-select, no branch
    aq1.y = 0.0f;

    const float* Kp = Kf + (size_t)b * HW * 8;      // [p][8]
    const float* Vb = Vf + b * Cdim * HW;           // [c][p]

    // ---------- pass 1: row maxima ----------
    float rmax[8];
#pragma unroll
    for (int r = 0; r < 8; ++r) rmax[r] = -3.0e38f;

    for (int t = 0; t < PTILES; ++t) {
        const int p = t * 16 + l16;
        const float2 kpair = *(const float2*)(Kp + (size_t)p * 8 + 2 * half);
        const float  k4    = Kp[(size_t)p * 8 + 4];
        v2f kb0, kb1;
        kb0.x = kpair.x;  kb0.y = kpair.y;
        kb1.x = (half == 0) ? k4 : 0.0f;
        kb1.y = 0.0f;
        v8f e = {};
        e = __builtin_amdgcn_wmma_f32_16x16x4_f32(false, aq0, false, kb0, (short)0, e, false, false);
        e = __builtin_amdgcn_wmma_f32_16x16x4_f32(false, aq1, false, kb1, (short)0, e, false, false);
#pragma unroll
        for (int r = 0; r < 8; ++r) rmax[r] = fmaxf(rmax[r], e[r]);
    }
    // butterfly max within each 16-lane half (C/D layout: row m=r+8*half, col=l16)
#pragma unroll
    for (int off = 1; off < 16; off <<= 1) {
#pragma unroll
        for (int r = 0; r < 8; ++r)
            rmax[r] = fmaxf(rmax[r], __shfl_xor(rmax[r], off, 32));
    }

    // ---------- pass 2: exp + out GEMM (col 5 of B is all-ones -> row sums) ----------
    const float L2E = 1.44269504088896340736f;
    const bool  cvalid = (l16 < Cdim);
    const float fillv  = (l16 == Cdim) ? 1.0f : 0.0f;   // ones column / zero columns
    const int   ceff   = cvalid ? l16 : 0;              // clamped (safe) V row
    v8f dacc = {};

    for (int t = 0; t < PTILES; ++t) {
        const int p = t * 16 + l16;
        const float2 kpair = *(const float2*)(Kp + (size_t)p * 8 + 2 * half);
        const float  k4    = Kp[(size_t)p * 8 + 4];
        v2f kb0, kb1;
        kb0.x = kpair.x;  kb0.y = kpair.y;
        kb1.x = (half == 0) ? k4 : 0.0f;
        kb1.y = 0.0f;
        v8f e = {};
        e = __builtin_amdgcn_wmma_f32_16x16x4_f32(false, aq0, false, kb0, (short)0, e, false, false);
        e = __builtin_amdgcn_wmma_f32_16x16x4_f32(false, aq1, false, kb1, (short)0, e, false, false);

        float tmp[8];
#pragma unroll
        for (int r = 0; r < 8; ++r) tmp[r] = exp2f((e[r] - rmax[r]) * L2E);
        // store exp tile as Etile[p_local][m]
        float* dst = &Etile[l16 * 16 + 8 * half];
        ((float4*)dst)[0] = make_float4(tmp[0], tmp[1], tmp[2], tmp[3]);
        ((float4*)dst)[1] = make_float4(tmp[4], tmp[5], tmp[6], tmp[7]);
        __syncthreads();

#pragma unroll
        for (int ch = 0; ch < 4; ++ch) {
            const int kl = ch * 4 + 2 * half;           // local p index for v0
            v2f ea, bv;
            ea.x = Etile[(kl + 0) * 16 + l16];          // A-layout read of exp tile
            ea.y = Etile[(kl + 1) * 16 + l16];
            const int gp = t * 16 + kl;                 // even -> 8B-aligned float2
            const float2 vv = *(const float2*)(Vb + ceff * HW + gp);
            bv.x = cvalid ? vv.x : fillv;               // value-select, no branch
            bv.y = cvalid ? vv.y : fillv;
            dacc = __builtin_amdgcn_wmma_f32_16x16x4_f32(false, ea, false, bv, (short)0, dacc, false, false);
        }
        __syncthreads();
    }

    // ---------- write back: y = gamma * (out/rowsum) + x ----------
    const float g = gamma[0];
#pragma unroll
    for (int r = 0; r < 8; ++r) {
        const float rsum = __shfl(dacc[r], half * 16 + 5, 32);  // col 5 = softmax denom
        const int m    = r + 8 * half;
        const int slot = tt * 16 + m;
        if (l16 < Cdim && slot < count) {
            const int n   = lists[b * Npts + slot];
            const int pix = idx_h[n] * Wdim + idx_w[n];
            const int addr = (b * Cdim + l16) * HW + pix;
            y[addr] = g * (dacc[r] / rsum) + x[addr];
        }
    }
}

extern "C" void kernel_launch(void* const* d_in, const int* in_sizes, int n_in,
                              void* d_out, int out_size, void* d_ws, size_t ws_size,
                              hipStream_t stream)
{
    (void)in_sizes; (void)n_in; (void)out_size; (void)ws_size;
    const float* x     = (const float*)d_in[0];
    // d_in[1] = x_teature (unused by the reference math)
    const float* qw    = (const float*)d_in[2];
    const float* qb    = (const float*)d_in[3];
    const float* kw    = (const float*)d_in[4];
    const float* kb    = (const float*)d_in[5];
    const float* vw    = (const float*)d_in[6];
    const float* vb    = (const float*)d_in[7];
    const float* gamma = (const float*)d_in[8];
    const int*   idx_b = (const int*)d_in[9];
    const int*   idx_h = (const int*)d_in[10];
    const int*   idx_w = (const int*)d_in[11];
    float* y = (float*)d_out;

    char* ws = (char*)d_ws;
    float* Kf   = (float*)ws; ws += (size_t)Bdim * HW * 8 * sizeof(float);   // p-major, padded
    float* Vf   = (float*)ws; ws += (size_t)Bdim * Cdim * HW * sizeof(float);
    float* Qg   = (float*)ws; ws += (size_t)Npts * 8 * sizeof(float);
    int*   cnts = (int*)ws;   ws += 64;                       // 4 counters + padding
    int*   lists= (int*)ws;   ws += (size_t)Bdim * Npts * sizeof(int);

    // y starts as a copy of x; attention kernel overwrites the gathered pixels
    hipMemcpyAsync(y, x, (size_t)Bdim * Cdim * HW * sizeof(float),
                   hipMemcpyDeviceToDevice, stream);

    qkv_map_kernel<<<(Bdim * HW + 255) / 256, 256, 0, stream>>>(x, kw, kb, vw, vb, Kf, Vf);
    q_gather_kernel<<<(Npts + 255) / 256, 256, 0, stream>>>(x, qw, qb, idx_b, idx_h, idx_w, Qg);
    bucket_init_kernel<<<1, 32, 0, stream>>>(cnts);
    bucket_kernel<<<(Npts + 255) / 256, 256, 0, stream>>>(idx_b, cnts, lists);
    attn_kernel<<<dim3(Bdim, PTILES), 32, 0, stream>>>(Kf, Vf, Qg, lists, cnts,
                                                       idx_h, idx_w, x, gamma, y);
}